// RMAC_42511586296006
// MI455X (gfx1250) — compile-verified
//
#include <hip/hip_runtime.h>
#include <hip/hip_bf16.h>
#include <stdint.h>

// ---------------------------------------------------------------------------
// RMAC pooling, MI455X (gfx1250, wave32).
// Memory-bound: 134MB read @ 23.3 TB/s ~ 5.8us floor. Data movement uses the
// CDNA5 async global->LDS path (ASYNCcnt) so the bulk copy bypasses VGPRs.
// ---------------------------------------------------------------------------

typedef int v4i_t __attribute__((ext_vector_type(4)));
typedef __attribute__((address_space(1))) v4i_t glb_v4i;   // global 128-bit
typedef __attribute__((address_space(3))) v4i_t lds_v4i;   // LDS 128-bit
typedef __attribute__((address_space(3))) void lds_void_t;

__device__ __forceinline__ void async_g2l_b128(void* lds_ptr, const void* gptr) {
#if defined(__has_builtin) && __has_builtin(__builtin_amdgcn_global_load_async_to_lds_b128)
  __builtin_amdgcn_global_load_async_to_lds_b128(
      (glb_v4i*)gptr, (lds_v4i*)lds_ptr, /*offset=*/0, /*cpol=*/0);
#else
  unsigned off = (unsigned)(uintptr_t)(lds_void_t*)lds_ptr;
  asm volatile("global_load_async_to_lds_b128 %0, %1, off"
               :: "v"(off), "v"(gptr) : "memory");
#endif
}

__device__ __forceinline__ void wait_asynccnt0() {
#if defined(__has_builtin) && __has_builtin(__builtin_amdgcn_s_wait_asynccnt)
  __builtin_amdgcn_s_wait_asynccnt(0);
#else
  asm volatile("s_wait_asynccnt 0" ::: "memory");
#endif
}

__device__ __forceinline__ void wait_dscnt0() {
#if defined(__has_builtin) && __has_builtin(__builtin_amdgcn_s_wait_dscnt)
  __builtin_amdgcn_s_wait_dscnt(0);
#else
  asm volatile("s_wait_dscnt 0" ::: "memory");
#endif
}

#define NEG_HUGE (-3.402823466e+38f)

// Stage 1: one wave per (b,c) plane (32x32 f32 = 4KB). 8 waves / 256-thr block.
// Async-copy plane into LDS, lane=column, 6 row-window running maxes, then
// lanes 0..13 reduce the 14 region maxes across column windows.
__global__ __launch_bounds__(256) void rmac_region_max(
    const float* __restrict__ x, float* __restrict__ M, int planes, int C) {
  __shared__ float plane[8][1024];     // 8 x 4KB
  __shared__ float cmax[8][32][6];     // per-column row-window maxes

  const int wave = threadIdx.x >> 5;
  const int lane = threadIdx.x & 31;
  const int p = blockIdx.x * 8 + wave;          // plane = b*C + c
  if (p >= planes) return;                      // whole wave exits together

  const float* gbase = x + (size_t)p * 1024;

  // 8 async b128 ops per wave: 32 lanes x 16B = 512B each -> 4KB total.
#pragma unroll
  for (int it = 0; it < 8; ++it) {
    async_g2l_b128(&plane[wave][it * 128 + lane * 4],
                   gbase + it * 128 + lane * 4);
  }
  wait_asynccnt0();
  __builtin_amdgcn_wave_barrier();

  // Row windows: w0=[0,32) w1=[0,21) w2=[11,32) w3=[0,16) w4=[8,24) w5=[16,32)
  float m0 = NEG_HUGE, m1 = NEG_HUGE, m2 = NEG_HUGE;
  float m3 = NEG_HUGE, m4 = NEG_HUGE, m5 = NEG_HUGE;
#pragma unroll
  for (int r = 0; r < 32; ++r) {
    float v = plane[wave][r * 32 + lane];       // conflict-free: 32 consecutive
    m0 = fmaxf(m0, v);
    if (r < 21)           m1 = fmaxf(m1, v);
    if (r >= 11)          m2 = fmaxf(m2, v);
    if (r < 16)           m3 = fmaxf(m3, v);
    if (r >= 8 && r < 24) m4 = fmaxf(m4, v);
    if (r >= 16)          m5 = fmaxf(m5, v);
  }
  cmax[wave][lane][0] = m0;
  cmax[wave][lane][1] = m1;
  cmax[wave][lane][2] = m2;
  cmax[wave][lane][3] = m3;
  cmax[wave][lane][4] = m4;
  cmax[wave][lane][5] = m5;
  __builtin_amdgcn_wave_barrier();
  wait_dscnt0();                                 // wave-synchronous LDS hand-off
  __builtin_amdgcn_wave_barrier();

  // 14 regions: lane 0 -> (w0,w0) [full image, weight 2 downstream],
  // lanes 1..4 -> {w1,w2}x{w1,w2}, lanes 5..13 -> {w3,w4,w5}x{w3,w4,w5}.
  if (lane < 14) {
    int rwi, cwi;
    if (lane == 0)      { rwi = 0; cwi = 0; }
    else if (lane < 5)  { rwi = 1 + ((lane - 1) >> 1); cwi = 1 + ((lane - 1) & 1); }
    else                { int t = lane - 5; rwi = 3 + t / 3; cwi = 3 + t % 3; }
    const int s = (cwi == 2) ? 11 : (cwi == 4) ? 8 : (cwi == 5) ? 16 : 0;
    const int n = (cwi == 0) ? 32 : (cwi < 3) ? 21 : 16;
    float m = NEG_HUGE;
    for (int t = 0; t < n; ++t) m = fmaxf(m, cmax[wave][s + t][rwi]);
    const int b = p / C, c = p - b * C;
    M[((size_t)b * 14 + lane) * C + c] = m;      // layout [b][region][c]
  }
}

// Stage 2: one wave per (b, region): sinv = 1/(||M[b,r,:]||_2 + 1e-6)
__global__ __launch_bounds__(32) void rmac_norms(
    const float* __restrict__ M, float* __restrict__ sinv, int BR, int C) {
  const int br = blockIdx.x;
  if (br >= BR) return;
  const int lane = threadIdx.x;
  const float* v = M + (size_t)br * C;
  float s = 0.0f;
  for (int i = lane; i < C; i += 32) { float t = v[i]; s += t * t; }
#pragma unroll
  for (int o = 16; o > 0; o >>= 1) s += __shfl_xor(s, o, 32);
  if (lane == 0) sinv[br] = 1.0f / (sqrtf(s) + 1e-6f);
}

// Stage 3: out[b,c] = sum_r weight_r * M[b,r,c] * sinv[b,r]; weight_0 = 2
// (global max-pool vector == l=1 region vector in the reference).
__global__ __launch_bounds__(256) void rmac_combine(
    const float* __restrict__ M, const float* __restrict__ sinv,
    float* __restrict__ out, int B, int C) {
  const int idx = blockIdx.x * blockDim.x + threadIdx.x;   // b*C + c
  if (idx >= B * C) return;
  const int b = idx / C, c = idx - b * C;
  const float* mb = M + (size_t)b * 14 * C + c;
  const float* sb = sinv + b * 14;
  float acc = 0.0f;
#pragma unroll
  for (int r = 0; r < 14; ++r) {
    const float w = (r == 0) ? 2.0f : 1.0f;
    acc += w * mb[(size_t)r * C] * sb[r];
  }
  out[idx] = acc;
}

extern "C" void kernel_launch(void* const* d_in, const int* in_sizes, int n_in,
                              void* d_out, int out_size, void* d_ws, size_t ws_size,
                              hipStream_t stream) {
  (void)n_in; (void)out_size; (void)ws_size;
  const float* x = (const float*)d_in[0];
  float* out = (float*)d_out;

  const int C = 512;
  const int planes = in_sizes[0] / 1024;   // B*C (H*W = 1024)
  const int B = planes / C;

  float* M    = (float*)d_ws;                       // B*14*C floats (~1.75MB)
  float* sinv = M + (size_t)B * 14 * C;             // B*14 floats

  rmac_region_max<<<(planes + 7) / 8, 256, 0, stream>>>(x, M, planes, C);
  rmac_norms<<<B * 14, 32, 0, stream>>>(M, sinv, B * 14, C);
  rmac_combine<<<(B * C + 255) / 256, 256, 0, stream>>>(M, sinv, out, B, C);
}